// ShortScaleTGN_23450521436438
// MI455X (gfx1250) — compile-verified
//
#include <hip/hip_runtime.h>
#include <math.h>

#define N_NODES 100000
#define N_EDGES 64
#define NODE_F  11
#define EDGE_F  30
#define MEMD    64
#define TIME_D  16
#define MSG_IN  174

typedef __attribute__((ext_vector_type(2))) float v2f;
typedef __attribute__((ext_vector_type(8))) float v8f;

// workspace layout (float offsets)
#define OFF_G      0        // 100000 gate logits
#define OFF_M      100000   // 1 global max
#define OFF_SUMEXP 100001   // 1
#define OFF_POOLED 100002   // 64 pooled accumulator
#define OFF_GVEC   100066   // 16 fused gate vector (11 real)
#define OFF_GBIAS  100082   // 1
#define OFF_WC     100083   // 16x64 fused proj weights (rows 11..15 zero)
#define OFF_CBIAS  101107   // 64
#define OFF_PATCHG 101171   // 128
#define OFF_UPD    101299   // 128x64 updated memory rows
#define OFF_IDS    109491   // 128 ints (via int* view)
#define OFF_CANON  109619   // 128 ints

// ---------------------------------------------------------------- K0: fuse weights
__global__ void k0_prep(const float* Wnp, const float* bnp, const float* Wgate,
                        const float* bgate, const float* Wproj, const float* bproj,
                        float* ws) {
  int t = threadIdx.x;
  for (int idx = t; idx < 16 * 64; idx += blockDim.x) {
    int k = idx >> 6, j = idx & 63;
    float v = 0.f;
    if (k < NODE_F)
      for (int m = 0; m < 64; ++m) v += Wnp[k * 64 + m] * Wproj[m * 64 + j];
    ws[OFF_WC + idx] = v;
  }
  for (int k = t; k < 16; k += blockDim.x) {
    float v = 0.f;
    if (k < NODE_F)
      for (int m = 0; m < 64; ++m) v += Wnp[k * 64 + m] * Wgate[m];
    ws[OFF_GVEC + k] = v;
  }
  for (int j = t; j < 64; j += blockDim.x) {
    float v = bproj[j];
    for (int m = 0; m < 64; ++m) v += bnp[m] * Wproj[m * 64 + j];
    ws[OFF_CBIAS + j] = v;
    ws[OFF_POOLED + j] = 0.f;
  }
  if (t == 0) {
    float v = bgate[0];
    for (int m = 0; m < 64; ++m) v += bnp[m] * Wgate[m];
    ws[OFF_GBIAS] = v;
    ws[OFF_SUMEXP] = 0.f;
  }
}

// ---------------------------------------------------------------- K1: sequential edge scan
__global__ __launch_bounds__(256) void k1_edges(
    const float* nf, const int* src, const int* dst, const float* ts, const float* ef,
    const float* Wnp, const float* bnp, const float* Wmsg, const float* bmsg,
    const float* Wih, const float* Whh, const float* bih, const float* bhh,
    const float* t2vw0, const float* t2vb0, const float* t2vW, const float* t2vB,
    const float* Wgate, const float* bgate, float* ws) {
  __shared__ float rows[128][64];
  __shared__ float te[64][16];
  __shared__ float xv[MSG_IN];
  __shared__ float msg[64];
  __shared__ float gi[192];
  __shared__ float gh[192];
  __shared__ int nodes[128];
  __shared__ int canon[128];
  int t = threadIdx.x;
  if (t < 128) nodes[t] = (t < 64) ? src[t] : dst[t - 64];
  __syncthreads();
  if (t < 128) {
    int c = t;
    for (int j = 0; j < t; ++j)
      if (nodes[j] == nodes[t]) { c = j; break; }
    canon[t] = c;
  }
  for (int idx = t; idx < 64 * 16; idx += blockDim.x) {
    int e = idx >> 4, c = idx & 15;
    float tv = ts[e];
    te[e][c] = (c == 0) ? (tv * t2vw0[0] + t2vb0[0])
                        : sinf(tv * t2vW[c - 1] + t2vB[c - 1]);
  }
  __syncthreads();
  // init memory rows for touched nodes: nf @ W_np + b_np
  for (int idx = t; idx < 128 * 64; idx += blockDim.x) {
    int i = idx >> 6, j = idx & 63;
    if (canon[i] == i) {
      const float* r = nf + (long)nodes[i] * NODE_F;
      float v = bnp[j];
      for (int k = 0; k < NODE_F; ++k) v += r[k] * Wnp[k * 64 + j];
      rows[i][j] = v;
    }
  }
  __syncthreads();
  for (int e = 0; e < N_EDGES; ++e) {
    int cs = canon[e], cd = canon[64 + e];
    if (t < MSG_IN) {
      float v;
      if (t < 64) v = rows[cs][t];
      else if (t < 128) v = rows[cd][t - 64];
      else if (t < 158) v = ef[e * EDGE_F + (t - 128)];
      else v = te[e][t - 158];
      xv[t] = v;
    }
    __syncthreads();
    if (t < 64) {
      float a = bmsg[t];
      for (int k = 0; k < MSG_IN; ++k) a += Wmsg[k * 64 + t] * xv[k];
      msg[t] = fmaxf(a, 0.f);
    }
    __syncthreads();
    if (t < 192) {
      float a = bih[t];
      for (int k = 0; k < 64; ++k) a += Wih[t * 64 + k] * msg[k];
      gi[t] = a;
    }
    __syncthreads();
    // src GRU then dst GRU (dst must observe src update when rows alias)
    for (int half = 0; half < 2; ++half) {
      int c = half ? cd : cs;
      if (t < 192) {
        float a = bhh[t];
        for (int k = 0; k < 64; ++k) a += Whh[t * 64 + k] * rows[c][k];
        gh[t] = a;
      }
      __syncthreads();
      float nv = 0.f;
      if (t < 64) {
        float r = 1.f / (1.f + expf(-(gi[t] + gh[t])));
        float z = 1.f / (1.f + expf(-(gi[64 + t] + gh[64 + t])));
        float nn = tanhf(gi[128 + t] + r * gh[128 + t]);
        nv = (1.f - z) * nn + z * rows[c][t];
      }
      __syncthreads();
      if (t < 64) rows[c][t] = nv;
      __syncthreads();
    }
  }
  int* iws = (int*)ws;
  if (t < 128) {
    iws[OFF_IDS + t] = nodes[t];
    iws[OFF_CANON + t] = canon[t];
    if (canon[t] == t) {
      float a = bgate[0];
      for (int j = 0; j < 64; ++j) a += rows[t][j] * Wgate[j];
      ws[OFF_PATCHG + t] = a;
    }
  }
  __syncthreads();
  for (int idx = t; idx < 128 * 64; idx += blockDim.x) {
    int i = idx >> 6;
    if (canon[i] == i) ws[OFF_UPD + idx] = rows[i][idx & 63];
  }
}

// ---------------------------------------------------------------- K2: gate logits (fused)
__global__ void k2_gate(const float* nf, float* ws) {
  int n = blockIdx.x * blockDim.x + threadIdx.x;
  if (n < N_NODES) {
    const float* r = nf + (long)n * NODE_F;
    float g = ws[OFF_GBIAS];
    for (int k = 0; k < NODE_F; ++k) g += r[k] * ws[OFF_GVEC + k];
    ws[OFF_G + n] = g;
  }
}

// ---------------------------------------------------------------- K2b: patch touched logits
__global__ void k2b_patch(float* ws) {
  int t = threadIdx.x;
  int* iws = (int*)ws;
  if (t < 128 && iws[OFF_CANON + t] == t)
    ws[OFF_G + iws[OFF_IDS + t]] = ws[OFF_PATCHG + t];
}

// ---------------------------------------------------------------- K3: global max
__global__ void k3_max(float* ws) {
  __shared__ float sm[1024];
  int t = threadIdx.x;
  float m = -3.4e38f;
  for (int n = t; n < N_NODES; n += blockDim.x) m = fmaxf(m, ws[OFF_G + n]);
  sm[t] = m;
  __syncthreads();
  for (int s = blockDim.x >> 1; s > 0; s >>= 1) {
    if (t < s) sm[t] = fmaxf(sm[t], sm[t + s]);
    __syncthreads();
  }
  if (t == 0) ws[OFF_M] = sm[0];
}

// ---------------------------------------------------------------- K4: WMMA pooled readout
__global__ __launch_bounds__(256) void k4_pool(const float* nf, float* ws) {
  __shared__ float As[8][16][16];  // per-wave 16x16 A tile (K padded 11->16)
  __shared__ float wl[8][16];      // softmax weights per node in tile
  __shared__ float pooled_lds[64];
  __shared__ float sum_lds;
  int t = threadIdx.x;
  int wave = t >> 5, lane = t & 31;
  int ml = lane & 15, kh = lane >> 4;
  if (t < 64) pooled_lds[t] = 0.f;
  if (t == 0) sum_lds = 0.f;
  float Mv = ws[OFF_M];
  // B fragments: Wc 16x64, 4 output tiles x 4 k-steps (invariant per block)
  v2f b[4][4];
#pragma unroll
  for (int t4 = 0; t4 < 4; ++t4)
#pragma unroll
    for (int s = 0; s < 4; ++s) {
      int k0 = 4 * s + 2 * kh;
      int j = t4 * 16 + ml;
      b[t4][s].x = ws[OFF_WC + k0 * 64 + j];
      b[t4][s].y = ws[OFF_WC + (k0 + 1) * 64 + j];
    }
  float cb[4];
#pragma unroll
  for (int t4 = 0; t4 < 4; ++t4) cb[t4] = ws[OFF_CBIAS + t4 * 16 + ml];
  float pacc[4] = {0.f, 0.f, 0.f, 0.f};
  float sexp = 0.f;
  int nspans = (N_NODES + 127) / 128;
  for (int span = blockIdx.x; span < nspans; span += gridDim.x) {
    long base = (long)span * 128;
    for (int idx = t; idx < 8 * 16 * 16; idx += 256) ((float*)As)[idx] = 0.f;
    __syncthreads();
    // coalesced stage of 128 contiguous nf rows (128*11 floats)
    for (int idx = t; idx < 128 * NODE_F; idx += 256) {
      int loc = idx / NODE_F, k = idx % NODE_F;
      long node = base + loc;
      if (node < N_NODES) As[loc >> 4][loc & 15][k] = nf[node * NODE_F + k];
    }
    if (t < 128) {
      long node = base + t;
      wl[t >> 4][t & 15] = (node < N_NODES) ? expf(ws[OFF_G + node] - Mv) : 0.f;
    }
    __syncthreads();
    v2f a[4];
#pragma unroll
    for (int s = 0; s < 4; ++s) {
      a[s].x = As[wave][ml][4 * s + 2 * kh];
      a[s].y = As[wave][ml][4 * s + 2 * kh + 1];
    }
    if (kh == 0) sexp += wl[wave][ml];
#pragma unroll
    for (int t4 = 0; t4 < 4; ++t4) {
      v8f acc = {0.f, 0.f, 0.f, 0.f, 0.f, 0.f, 0.f, 0.f};
#pragma unroll
      for (int s = 0; s < 4; ++s)
        acc = __builtin_amdgcn_wmma_f32_16x16x4_f32(
            false, a[s], false, b[t4][s], (short)0, acc, false, false);
#pragma unroll
      for (int r = 0; r < 8; ++r) {
        float w = wl[wave][8 * kh + r];
        pacc[t4] += w * (acc[r] + cb[t4]);
      }
    }
    __syncthreads();
  }
#pragma unroll
  for (int t4 = 0; t4 < 4; ++t4) atomicAdd(&pooled_lds[t4 * 16 + ml], pacc[t4]);
  if (kh == 0) atomicAdd(&sum_lds, sexp);
  __syncthreads();
  if (t < 64) atomicAdd(&ws[OFF_POOLED + t], pooled_lds[t]);
  if (t == 0) atomicAdd(&ws[OFF_SUMEXP], sum_lds);
}

// ---------------------------------------------------------------- K5: correct touched nodes
__global__ void k5_patch(const float* nf, const float* Wproj, const float* bproj,
                         float* ws) {
  int t = threadIdx.x;
  int j = t & 63, grp = t >> 6;
  int* iws = (int*)ws;
  float Mv = ws[OFF_M];
  for (int i = grp; i < 128; i += 4) {
    if (iws[OFF_CANON + i] != i) continue;
    int id = iws[OFF_IDS + i];
    float w = expf(ws[OFF_G + id] - Mv);
    float pu = bproj[j];
    for (int m = 0; m < 64; ++m) pu += ws[OFF_UPD + i * 64 + m] * Wproj[m * 64 + j];
    float po = ws[OFF_CBIAS + j];
    const float* r = nf + (long)id * NODE_F;
    for (int k = 0; k < NODE_F; ++k) po += r[k] * ws[OFF_WC + k * 64 + j];
    atomicAdd(&ws[OFF_POOLED + j], w * (pu - po));
  }
}

// ---------------------------------------------------------------- K6: normalize
__global__ void k6_final(const float* ws, float* out) {
  int j = threadIdx.x;
  if (j < 64) out[j] = ws[OFF_POOLED + j] / ws[OFF_SUMEXP];
}

extern "C" void kernel_launch(void* const* d_in, const int* in_sizes, int n_in,
                              void* d_out, int out_size, void* d_ws, size_t ws_size,
                              hipStream_t stream) {
  const float* nf = (const float*)d_in[0];
  const int* src = (const int*)d_in[1];
  const int* dst = (const int*)d_in[2];
  const float* ts = (const float*)d_in[3];
  const float* ef = (const float*)d_in[4];
  const float* Wnp = (const float*)d_in[5];
  const float* bnp = (const float*)d_in[6];
  const float* Wmsg = (const float*)d_in[7];
  const float* bmsg = (const float*)d_in[8];
  const float* Wih = (const float*)d_in[9];
  const float* Whh = (const float*)d_in[10];
  const float* bih = (const float*)d_in[11];
  const float* bhh = (const float*)d_in[12];
  const float* t2vw0 = (const float*)d_in[13];
  const float* t2vb0 = (const float*)d_in[14];
  const float* t2vW = (const float*)d_in[15];
  const float* t2vB = (const float*)d_in[16];
  const float* Wgate = (const float*)d_in[17];
  const float* bgate = (const float*)d_in[18];
  const float* Wproj = (const float*)d_in[19];
  const float* bproj = (const float*)d_in[20];
  float* ws = (float*)d_ws;
  float* out = (float*)d_out;

  k0_prep<<<1, 256, 0, stream>>>(Wnp, bnp, Wgate, bgate, Wproj, bproj, ws);
  k1_edges<<<1, 256, 0, stream>>>(nf, src, dst, ts, ef, Wnp, bnp, Wmsg, bmsg,
                                  Wih, Whh, bih, bhh, t2vw0, t2vb0, t2vW, t2vB,
                                  Wgate, bgate, ws);
  k2_gate<<<(N_NODES + 255) / 256, 256, 0, stream>>>(nf, ws);
  k2b_patch<<<1, 128, 0, stream>>>(ws);
  k3_max<<<1, 1024, 0, stream>>>(ws);
  k4_pool<<<160, 256, 0, stream>>>(nf, ws);
  k5_patch<<<1, 256, 0, stream>>>(nf, Wproj, bproj, ws);
  k6_final<<<1, 64, 0, stream>>>(ws, out);
}